// Router_43276090474708
// MI455X (gfx1250) — compile-verified
//
#include <hip/hip_runtime.h>

#define SEQ  16384
#define DIM  4096
#define NEXP 16
#define ROWSTRIDE 36   // 32 floats + 4 pad: keeps 16B LDS alignment, spreads banks

typedef __attribute__((ext_vector_type(2))) float v2f;
typedef __attribute__((ext_vector_type(8))) float v8f;

// ---------------------------------------------------------------------------
// Kernel 1: router logits = x @ W + b  via V_WMMA_F32_16X16X4_F32
// One wave owns a 16-row tile of x; x is streamed exactly once (268 MB).
// ---------------------------------------------------------------------------
__global__ __launch_bounds__(256)
void router_gemm(const float* __restrict__ x, const float* __restrict__ W,
                 const float* __restrict__ bias, float* __restrict__ logits) {
  __shared__ float tileA[8 * 16 * ROWSTRIDE];

  const int lane   = threadIdx.x & 31;
  const int wave   = threadIdx.x >> 5;
  const int laneHi = lane >> 4;     // 0: K pair (k,k+1), 1: K pair (k+2,k+3)
  const int laneLo = lane & 15;     // row (A) / column N (B,C)
  const int rowBase = (blockIdx.x * 8 + wave) * 16;
  float* myTile = tileA + wave * (16 * ROWSTRIDE);

  v8f c = {};

  for (int kbase = 0; kbase < DIM; kbase += 32) {
    // Coalesced stage of a 16x32 f32 tile of x into LDS (4 x 512B bursts).
#pragma unroll
    for (int i = 0; i < 4; ++i) {
      int idx = i * 32 + lane;
      int r   = idx >> 3;          // 0..15
      int cg  = (idx & 7) << 2;    // 0,4,...,28
      const float4 v = *(const float4*)(x + (size_t)(rowBase + r) * DIM + kbase + cg);
      *(float4*)(myTile + r * ROWSTRIDE + cg) = v;
    }
    if (kbase + 32 < DIM)
      __builtin_prefetch(x + (size_t)(rowBase + (lane >> 1)) * DIM + kbase + 32, 0, 3);

    // 8 WMMA steps of K=4 each over the staged tile.
#pragma unroll
    for (int s = 0; s < 8; ++s) {
      int k2 = s * 4 + 2 * laneHi;
      // A fragment: 16x4 f32 per ISA layout (VGPR0=K even, VGPR1=K odd)
      v2f a = *(const v2f*)(myTile + laneLo * ROWSTRIDE + k2);
      // B fragment: 4x16 f32, lanes = N, halves = K pairs. W is [DIM, NEXP].
      int kk = kbase + k2;
      v2f b;
      b.x = W[kk * NEXP + laneLo];
      b.y = W[kk * NEXP + NEXP + laneLo];
      c = __builtin_amdgcn_wmma_f32_16x16x4_f32(
          /*neg_a=*/false, a, /*neg_b=*/false, b,
          /*c_mod=*/(short)0, c, /*reuse_a=*/false, /*reuse_b=*/false);
    }
  }

  // C/D layout: VGPR v -> (M=v, N=lane) lanes 0-15; (M=v+8, N=lane-16) lanes 16-31.
  float bv = bias[laneLo];
  int m0 = laneHi * 8;
#pragma unroll
  for (int v = 0; v < 8; ++v)
    logits[(size_t)(rowBase + m0 + v) * NEXP + laneLo] = c[v] + bv;
}

// ---------------------------------------------------------------------------
// Kernel 2: top-2 (lax.top_k tie-break: lowest index), softmax-of-2,
// one-hot expert mask, capacity scalar.
// ---------------------------------------------------------------------------
__global__ __launch_bounds__(256)
void router_topk(const float* __restrict__ logits, float* __restrict__ out) {
  const int OFF_IDX = SEQ * NEXP;             // 262144
  const int OFF_W   = OFF_IDX + SEQ * 2;      // 294912
  const int OFF_M   = OFF_W + SEQ * 2;        // 327680
  const int OFF_CAP = OFF_M + SEQ * NEXP;     // 589824

  int s = blockIdx.x * blockDim.x + threadIdx.x;
  if (s >= SEQ) return;

  float l[NEXP];
  const float4* p = (const float4*)(logits + (size_t)s * NEXP);
#pragma unroll
  for (int i = 0; i < 4; ++i) {
    float4 v = p[i];
    l[i*4+0] = v.x; l[i*4+1] = v.y; l[i*4+2] = v.z; l[i*4+3] = v.w;
  }

  // argmax (strict > keeps lowest index on ties, matching lax.top_k)
  int i0 = 0; float v0 = l[0];
#pragma unroll
  for (int e = 1; e < NEXP; ++e)
    if (l[e] > v0) { v0 = l[e]; i0 = e; }
  // second max excluding i0
  int i1 = -1; float v1 = -3.402823466e38f;
#pragma unroll
  for (int e = 0; e < NEXP; ++e)
    if (e != i0 && l[e] > v1) { v1 = l[e]; i1 = e; }

  // softmax over [v0, v1]; v0 >= v1 so shift by v0
  float e1 = expf(v1 - v0);
  float inv = 1.0f / (1.0f + e1);
  float w0 = inv;
  float w1 = e1 * inv;

  out[OFF_IDX + s*2 + 0] = (float)i0;
  out[OFF_IDX + s*2 + 1] = (float)i1;
  out[OFF_W   + s*2 + 0] = w0;
  out[OFF_W   + s*2 + 1] = w1;

  // branch-free one-hot mask, written as float4s
  float4* mp = (float4*)(out + OFF_M + (size_t)s * NEXP);
#pragma unroll
  for (int i = 0; i < 4; ++i) {
    float4 mv;
    int e = i * 4;
    mv.x = (e+0 == i0 || e+0 == i1) ? 1.0f : 0.0f;
    mv.y = (e+1 == i0 || e+1 == i1) ? 1.0f : 0.0f;
    mv.z = (e+2 == i0 || e+2 == i1) ? 1.0f : 0.0f;
    mv.w = (e+3 == i0 || e+3 == i1) ? 1.0f : 0.0f;
    mp[i] = mv;
  }

  if (s == 0) out[OFF_CAP] = 2048.0f;  // min(16384, int(1.0*16384/16*2))
}

// ---------------------------------------------------------------------------
extern "C" void kernel_launch(void* const* d_in, const int* in_sizes, int n_in,
                              void* d_out, int out_size, void* d_ws, size_t ws_size,
                              hipStream_t stream) {
  const float* x    = (const float*)d_in[0];   // [SEQ, DIM]
  const float* W    = (const float*)d_in[1];   // [DIM, NEXP]
  const float* bias = (const float*)d_in[2];   // [NEXP]
  float* out = (float*)d_out;

  router_gemm<<<SEQ / 128, 256, 0, stream>>>(x, W, bias, out);
  router_topk<<<SEQ / 256, 256, 0, stream>>>(out, out);
}